// EGCL_28046136443386
// MI455X (gfx1250) — compile-verified
//
#include <hip/hip_runtime.h>

typedef __attribute__((ext_vector_type(2))) float v2f;
typedef __attribute__((ext_vector_type(8))) float v8f;

#define EGCL_EPS 1e-6f
#define EGCL_INV_SQRT2 0.70710678118654752f

// Cl(3,0) blade order [1,e1,e2,e3,e12,e13,e23,e123]; masks; idx<->mask is the 3/4 swap (involution)
constexpr int kMask[8]  = {0, 1, 2, 4, 3, 5, 6, 7};
constexpr int kGrade[8] = {0, 1, 1, 1, 2, 2, 2, 3};

static __device__ inline float sigmoidf_(float t) { return 1.f / (1.f + __expf(-t)); }

// D(16x16,f32) += A(16x4,f32) * B(4x16,f32)  — fp32 WMMA, chained over K
static __device__ inline v8f wmma4(v2f a, v2f b, v8f c) {
  return __builtin_amdgcn_wmma_f32_16x16x4_f32(false, a, false, b, (short)0, c, false, false);
}

// A fragment: grade-indexed weight W[COUT][CIN][4]; A(M,K) = W[M][K][g]; rows M>=COUT are zero.
// Layout (ISA 7.12.2, 32-bit A 16x4): lanes0-15 M, v0:K=k0+0 v1:K=k0+1; lanes16-31 v0:K=k0+2 v1:K=k0+3
template<int CIN, int COUT>
static __device__ inline v2f load_A(const float* __restrict__ W, int g, int k0, int lane) {
  const int M  = lane & 15;
  const int kb = k0 + ((lane >> 4) << 1);
  v2f a;
  if (M < COUT) {
    a[0] = W[(M * CIN + kb) * 4 + g];
    a[1] = W[(M * CIN + kb + 1) * 4 + g];
  } else {
    a[0] = 0.f; a[1] = 0.f;
  }
  return a;
}

// B fragment from per-wave LDS tile X[ch][blade][16]: B(K,N) = X[K][blade][N]
static __device__ inline v2f load_B(const float* __restrict__ X, int blade, int k0, int lane) {
  const int N  = lane & 15;
  const int kb = k0 + ((lane >> 4) << 1);
  v2f b;
  b[0] = X[(kb * 8 + blade) * 16 + N];
  b[1] = X[((kb + 1) * 8 + blade) * 16 + N];
  return b;
}

struct BlockW {
  const float *lw, *lb, *sa, *sb, *wl, *bl, *wr, *na, *gw, *ln;
};

// One CEGNN block: x = silu(lin(in)); out = layernorm((lin_l(x) + geo(x, norm(lin_r(x))))/sqrt2)
// In/out through the wave-private LDS tile X ([max16 ch][8 blades][16 items]).
template<int CIN, int COUT>
static __device__ void cemlp_block(const BlockW w, float* __restrict__ X, int lane) {
  const int half = lane >> 4;
  const int N    = lane & 15;

  // ---- mv_linear(in, lw) per blade via fp32 WMMA chain ----
  v8f acc[8];
#pragma unroll
  for (int i = 0; i < 8; ++i) {
    v8f c = {0.f, 0.f, 0.f, 0.f, 0.f, 0.f, 0.f, 0.f};
#pragma unroll
    for (int k0 = 0; k0 < CIN; k0 += 4)
      c = wmma4(load_A<CIN, COUT>(w.lw, kGrade[i], k0, lane), load_B(X, i, k0, lane), c);
    acc[i] = c;
  }
  // ---- bias (scalar blade) + mv_silu, lane-local ----
#pragma unroll
  for (int r = 0; r < 8; ++r) {
    const int M = r + half * 8;
    if (M < COUT) {
      acc[0][r] += w.lb[M];
      const float x0 = acc[0][r];
      const float q1 = acc[1][r]*acc[1][r] + acc[2][r]*acc[2][r] + acc[3][r]*acc[3][r];
      const float q2 = acc[4][r]*acc[4][r] + acc[5][r]*acc[5][r] + acc[6][r]*acc[6][r];
      const float q3 = acc[7][r]*acc[7][r];
      const float inv[4] = {x0, q1, q2, q3};
      float gate[4];
#pragma unroll
      for (int g = 0; g < 4; ++g)
        gate[g] = sigmoidf_(w.sa[M * 4 + g] * inv[g] + w.sb[M * 4 + g]);
#pragma unroll
      for (int i = 0; i < 8; ++i) acc[i][r] *= gate[kGrade[i]];
    }
  }

  // ---- stage x to LDS in B layout for the two sgp GEMMs ----
  __builtin_amdgcn_wave_barrier();
#pragma unroll
  for (int i = 0; i < 8; ++i)
#pragma unroll
    for (int r = 0; r < 8; ++r) {
      const int M = r + half * 8;
      if (M < COUT) X[(M * 8 + i) * 16 + N] = acc[i][r];
    }
  __builtin_amdgcn_wave_barrier();

  // ---- right = grade_normalize(mv_linear(x, wr), na) ----
  v8f racc[8];
#pragma unroll
  for (int i = 0; i < 8; ++i) {
    v8f c = {0.f, 0.f, 0.f, 0.f, 0.f, 0.f, 0.f, 0.f};
#pragma unroll
    for (int k0 = 0; k0 < COUT; k0 += 4)
      c = wmma4(load_A<COUT, COUT>(w.wr, kGrade[i], k0, lane), load_B(X, i, k0, lane), c);
    racc[i] = c;
  }
#pragma unroll
  for (int r = 0; r < 8; ++r) {
    const int M = r + half * 8;
    if (M < COUT) {
      float nq[4];
      nq[0] = fabsf(racc[0][r]);
      nq[1] = sqrtf(racc[1][r]*racc[1][r] + racc[2][r]*racc[2][r] + racc[3][r]*racc[3][r]);
      nq[2] = sqrtf(racc[4][r]*racc[4][r] + racc[5][r]*racc[5][r] + racc[6][r]*racc[6][r]);
      nq[3] = fabsf(racc[7][r]);
      float nn[4];
#pragma unroll
      for (int g = 0; g < 4; ++g)
        nn[g] = sigmoidf_(w.na[M * 4 + g]) * (nq[g] - 1.f) + 1.f;
#pragma unroll
      for (int i = 0; i < 8; ++i) racc[i][r] /= (nn[kGrade[i]] + EGCL_EPS);
    }
  }

  // ---- mv_linear(x, wl, bl) ----
  v8f lacc[8];
#pragma unroll
  for (int i = 0; i < 8; ++i) {
    v8f c = {0.f, 0.f, 0.f, 0.f, 0.f, 0.f, 0.f, 0.f};
#pragma unroll
    for (int k0 = 0; k0 < COUT; k0 += 4)
      c = wmma4(load_A<COUT, COUT>(w.wl, kGrade[i], k0, lane), load_B(X, i, k0, lane), c);
    lacc[i] = c;
  }
  // ---- weighted geometric product (64 nonzero Cayley terms, compile-time folded) ----
#pragma unroll
  for (int r = 0; r < 8; ++r) {
    const int M = r + half * 8;
    if (M < COUT) {
      lacc[0][r] += w.bl[M];
      const float* gwc = w.gw + M * 64;
#pragma unroll
      for (int ai = 0; ai < 8; ++ai) {
#pragma unroll
        for (int bi = 0; bi < 8; ++bi) {
          const int am  = kMask[ai], bm = kMask[bi];
          const int sgn = (__builtin_popcount((am >> 1) & bm) +
                           __builtin_popcount((am >> 2) & bm)) & 1;
          const int ki  = kMask[am ^ bm];
          const float t = acc[ai][r] * racc[bi][r] *
                          gwc[kGrade[ai] * 16 + kGrade[bi] * 4 + kGrade[ki]];
          lacc[ki][r] += sgn ? -t : t;
        }
      }
    }
  }

  // ---- /sqrt(2) + mv_layernorm (channel mean via lane-half swap) ----
  float csum = 0.f;
#pragma unroll
  for (int r = 0; r < 8; ++r) {
    float s2 = 0.f;
#pragma unroll
    for (int i = 0; i < 8; ++i) {
      lacc[i][r] *= EGCL_INV_SQRT2;
      s2 += lacc[i][r] * lacc[i][r];
    }
    csum += sqrtf(s2);
  }
  csum += __shfl_xor(csum, 16, 32);
  const float mscale = 1.f / (csum / (float)COUT + EGCL_EPS);
#pragma unroll
  for (int r = 0; r < 8; ++r) {
    const int M = r + half * 8;
    const float s = (M < COUT) ? (w.ln[M] * mscale) : 0.f;
#pragma unroll
    for (int i = 0; i < 8; ++i) lacc[i][r] *= s;
  }

  // ---- write block output back to LDS tile ----
  __builtin_amdgcn_wave_barrier();
#pragma unroll
  for (int i = 0; i < 8; ++i)
#pragma unroll
    for (int r = 0; r < 8; ++r) {
      const int M = r + half * 8;
      if (M < COUT) X[(M * 8 + i) * 16 + N] = lacc[i][r];
    }
  __builtin_amdgcn_wave_barrier();
}

__global__ void egcl_zero_kernel(float* p, int n) {
  int i = blockIdx.x * blockDim.x + threadIdx.x;
  if (i < n) p[i] = 0.f;
}

__global__ __launch_bounds__(128)
void egcl_edge_kernel(const float* __restrict__ h, const int* __restrict__ ei,
                      BlockW b0, BlockW b1,
                      float* __restrict__ hsum, float* __restrict__ cnt, int E) {
  __shared__ float smem[4][16 * 8 * 16];
  const int wave = threadIdx.x >> 5, lane = threadIdx.x & 31;
  const int tile = blockIdx.x * 4 + wave;
  if (tile * 16 >= E) return;
  float* X = &smem[wave][0];

  const int e    = lane & 15;
  const int half = lane >> 4;
  int eg = tile * 16 + e;
  if (eg >= E) eg = E - 1;               // safety clamp (E is a multiple of 16)
  const int row = ei[eg], col = ei[E + eg];
  if (half == 0) atomicAdd(&cnt[row], 1.f);
  __builtin_prefetch(&hsum[row * 64], 0, 3);   // global_prefetch_b8, scatter target into L2

  // x = h[row] - h[col] staged into B layout: X[c][i][e]
  const float4* hr = (const float4*)(h + (size_t)row * 64);
  const float4* hc = (const float4*)(h + (size_t)col * 64);
#pragma unroll
  for (int cc = 0; cc < 4; ++cc) {
    const int c = half * 4 + cc;
    float4 a0 = hr[c * 2], a1 = hr[c * 2 + 1];
    float4 q0 = hc[c * 2], q1 = hc[c * 2 + 1];
    float d[8] = {a0.x - q0.x, a0.y - q0.y, a0.z - q0.z, a0.w - q0.w,
                  a1.x - q1.x, a1.y - q1.y, a1.z - q1.z, a1.w - q1.w};
#pragma unroll
    for (int i = 0; i < 8; ++i) X[(c * 8 + i) * 16 + e] = d[i];
  }
  __builtin_amdgcn_wave_barrier();

  cemlp_block<8, 16>(b0, X, lane);
  cemlp_block<16, 8>(b1, X, lane);

  // segment-sum scatter (L2-resident accumulator)
#pragma unroll
  for (int cc = 0; cc < 4; ++cc) {
    const int c = half * 4 + cc;
#pragma unroll
    for (int i = 0; i < 8; ++i)
      atomicAdd(&hsum[(size_t)row * 64 + c * 8 + i], X[(c * 8 + i) * 16 + e]);
  }
}

__global__ __launch_bounds__(128)
void egcl_node_kernel(const float* __restrict__ h, const float* __restrict__ hsum,
                      const float* __restrict__ cnt,
                      BlockW b0, BlockW b1, float* __restrict__ out, int N) {
  __shared__ float smem[4][16 * 8 * 16];
  const int wave = threadIdx.x >> 5, lane = threadIdx.x & 31;
  const int tile = blockIdx.x * 4 + wave;
  if (tile * 16 >= N) return;
  float* X = &smem[wave][0];

  const int v    = lane & 15;
  const int half = lane >> 4;
  int n = tile * 16 + v;
  if (n >= N) n = N - 1;                 // safety clamp (N is a multiple of 16)
  const float invc = 1.f / fmaxf(cnt[n], 1.f);

  // input = concat([h, h_agg]) staged into B layout
#pragma unroll
  for (int cc = 0; cc < 4; ++cc) {
    const int c = half * 4 + cc;
#pragma unroll
    for (int i = 0; i < 8; ++i) {
      X[(c * 8 + i) * 16 + v]       = h[(size_t)n * 64 + c * 8 + i];
      X[((c + 8) * 8 + i) * 16 + v] = hsum[(size_t)n * 64 + c * 8 + i] * invc;
    }
  }
  __builtin_amdgcn_wave_barrier();

  cemlp_block<16, 16>(b0, X, lane);
  cemlp_block<16, 8>(b1, X, lane);

  // residual
#pragma unroll
  for (int cc = 0; cc < 4; ++cc) {
    const int c = half * 4 + cc;
#pragma unroll
    for (int i = 0; i < 8; ++i)
      out[(size_t)n * 64 + c * 8 + i] =
          h[(size_t)n * 64 + c * 8 + i] + X[(c * 8 + i) * 16 + v];
  }
}

extern "C" void kernel_launch(void* const* d_in, const int* in_sizes, int n_in,
                              void* d_out, int out_size, void* d_ws, size_t ws_size,
                              hipStream_t stream) {
  const float* h  = (const float*)d_in[0];
  const int*   ei = (const int*)d_in[1];
  const int N = in_sizes[0] / 64;   // [N,8,8]
  const int E = in_sizes[1] / 2;    // [2,E]

  // params flattened in setup_inputs() dict order:
  // edge[0], edge[1], node[0], node[1]; each block: lin_w, lin_b, silu_a, silu_b,
  // sgp_wl, sgp_bl, sgp_wr, sgp_na, sgp_gw, ln_a
  BlockW B[4];
  int p = 2;
  for (int b = 0; b < 4; ++b) {
    B[b].lw = (const float*)d_in[p++];
    B[b].lb = (const float*)d_in[p++];
    B[b].sa = (const float*)d_in[p++];
    B[b].sb = (const float*)d_in[p++];
    B[b].wl = (const float*)d_in[p++];
    B[b].bl = (const float*)d_in[p++];
    B[b].wr = (const float*)d_in[p++];
    B[b].na = (const float*)d_in[p++];
    B[b].gw = (const float*)d_in[p++];
    B[b].ln = (const float*)d_in[p++];
  }

  float* hsum = (float*)d_ws;             // [N,8,8] segment sums
  float* cnt  = hsum + (size_t)N * 64;    // [N]   segment counts
  const int zn = N * 65;
  egcl_zero_kernel<<<(zn + 255) / 256, 256, 0, stream>>>(hsum, zn);

  const int etiles = (E + 15) / 16;
  egcl_edge_kernel<<<(etiles + 3) / 4, 128, 0, stream>>>(h, ei, B[0], B[1], hsum, cnt, E);

  const int ntiles = (N + 15) / 16;
  egcl_node_kernel<<<(ntiles + 3) / 4, 128, 0, stream>>>(h, hsum, cnt, B[2], B[3],
                                                         (float*)d_out, N);
}